// AttentionBasedClassifier1_60928406061495
// MI455X (gfx1250) — compile-verified
//
#include <hip/hip_runtime.h>
#include <hip/hip_bf16.h>
#include <math.h>

typedef __attribute__((ext_vector_type(2))) float v2f;
typedef __attribute__((ext_vector_type(8))) float v8f;

#define BATCH 64
#define PLEN  128
#define DIM   1024

// One wave (32 lanes) per 16-row tile of one batch.
// Computes s[b,p] = c[b,p]·W_att and t[b,p] = c[b,p]·W_fc via
// V_WMMA_F32_16X16X4_F32, where c = concat(h,q) along K.
// A fragment (16x4 f32): lanes 0-15 hold rows M=0..15 at K=k,k+1 (v0,v1);
// lanes 16-31 hold the same rows at K=k+2,k+3.
// B fragment (4x16 f32): lane N = lane&15 holds column N; column 0 = W_att,
// column 1 = W_fc, columns 2..15 masked to zero (mask-multiply keeps EXEC
// uniform, as WMMA requires EXEC == all ones).
__global__ void __launch_bounds__(32)
att_score_kernel(const float* __restrict__ h,
                 const float* __restrict__ q,
                 const float* __restrict__ W_att,
                 const float* __restrict__ W_fc,
                 float* __restrict__ s_ws,
                 float* __restrict__ t_ws)
{
    const int b       = blockIdx.y;       // batch
    const int tile    = blockIdx.x;       // 0..7, 16 rows each
    const int lane    = threadIdx.x;      // 0..31 (wave32)
    const int n       = lane & 15;        // N column / row-in-tile
    const int hi      = lane >> 4;        // 0: K=+0,+1   1: K=+2,+3
    const int kOff    = hi * 2;
    const int rowBase = tile * 16;
    const int row     = rowBase + n;

    const float* __restrict__ hrow = h + ((size_t)b * PLEN + row) * DIM;
    const float* __restrict__ qrow = q + ((size_t)b * PLEN + row) * DIM;
    const float* __restrict__ wl   = (n == 1) ? W_fc : W_att; // n>=2 reads W_att harmlessly
    const float  msk = (n < 2) ? 1.0f : 0.0f;

    v8f acc0 = {}; v8f acc1 = {}; v8f acc2 = {}; v8f acc3 = {};

    // pass 0: h-part (K = 0..1023 of concat), pass 1: q-part (K = 1024..2047)
    for (int pass = 0; pass < 2; ++pass) {
        const float* __restrict__ arow = pass ? qrow : hrow;
        const float* __restrict__ wrow = wl + pass * DIM;
        for (int k = 0; k < DIM; k += 16) {
            v2f a0 = *(const v2f*)(arow + k +  0 + kOff);
            v2f a1 = *(const v2f*)(arow + k +  4 + kOff);
            v2f a2 = *(const v2f*)(arow + k +  8 + kOff);
            v2f a3 = *(const v2f*)(arow + k + 12 + kOff);
            v2f b0 = *(const v2f*)(wrow + k +  0 + kOff); b0 *= msk;
            v2f b1 = *(const v2f*)(wrow + k +  4 + kOff); b1 *= msk;
            v2f b2 = *(const v2f*)(wrow + k +  8 + kOff); b2 *= msk;
            v2f b3 = *(const v2f*)(wrow + k + 12 + kOff); b3 *= msk;
            // 8 args: (neg_a, A, neg_b, B, c_mod, C, reuse_a, reuse_b)
            acc0 = __builtin_amdgcn_wmma_f32_16x16x4_f32(false, a0, false, b0,
                                                         (short)0, acc0, false, false);
            acc1 = __builtin_amdgcn_wmma_f32_16x16x4_f32(false, a1, false, b1,
                                                         (short)0, acc1, false, false);
            acc2 = __builtin_amdgcn_wmma_f32_16x16x4_f32(false, a2, false, b2,
                                                         (short)0, acc2, false, false);
            acc3 = __builtin_amdgcn_wmma_f32_16x16x4_f32(false, a3, false, b3,
                                                         (short)0, acc3, false, false);
        }
    }

    v8f acc = (acc0 + acc1) + (acc2 + acc3);

    // D layout: lanes 0-15 -> (M = r, N = lane); lanes 16-31 -> (M = r+8, N = lane-16).
    // Column 0 = s, column 1 = t.
    if (n < 2) {
        float* dst = ((n == 0) ? s_ws : t_ws) + b * PLEN + rowBase + hi * 8;
        #pragma unroll
        for (int r = 0; r < 8; ++r) dst[r] = acc[r];
    }
}

// One block per batch: softmax over P=128 scores + weighted sum of t.
__global__ void __launch_bounds__(PLEN)
att_softmax_pool_kernel(const float* __restrict__ s_ws,
                        const float* __restrict__ t_ws,
                        const float* __restrict__ b_fc,
                        float* __restrict__ out)
{
    const int b = blockIdx.x;
    const int p = threadIdx.x;
    __shared__ float sh[PLEN];

    const float sv = s_ws[b * PLEN + p];
    const float tv = t_ws[b * PLEN + p];

    // max reduce
    sh[p] = sv;
    __syncthreads();
    for (int off = PLEN / 2; off > 0; off >>= 1) {
        if (p < off) sh[p] = fmaxf(sh[p], sh[p + off]);
        __syncthreads();
    }
    const float mx = sh[0];
    __syncthreads();

    const float e = expf(sv - mx);

    // sum(exp)
    sh[p] = e;
    __syncthreads();
    for (int off = PLEN / 2; off > 0; off >>= 1) {
        if (p < off) sh[p] += sh[p + off];
        __syncthreads();
    }
    const float denom = sh[0];
    __syncthreads();

    // sum(exp * t)
    sh[p] = e * tv;
    __syncthreads();
    for (int off = PLEN / 2; off > 0; off >>= 1) {
        if (p < off) sh[p] += sh[p + off];
        __syncthreads();
    }

    if (p == 0) out[b] = sh[0] / denom + b_fc[0];
}

extern "C" void kernel_launch(void* const* d_in, const int* in_sizes, int n_in,
                              void* d_out, int out_size, void* d_ws, size_t ws_size,
                              hipStream_t stream) {
    (void)in_sizes; (void)n_in; (void)out_size; (void)ws_size;
    const float* h     = (const float*)d_in[0]; // (B,P,D)
    const float* q     = (const float*)d_in[1]; // (B,P,D)
    const float* W_att = (const float*)d_in[2]; // (2D,1)
    // d_in[3] = b_att: softmax is shift-invariant -> mathematically a no-op
    const float* W_fc  = (const float*)d_in[4]; // (2D,1)
    const float* b_fc  = (const float*)d_in[5]; // (1,)
    float* out = (float*)d_out;                 // (B,1) fp32

    float* ws   = (float*)d_ws;
    float* s_ws = ws;                 // B*P floats
    float* t_ws = ws + BATCH * PLEN;  // B*P floats  (total 64 KB)

    dim3 grid(PLEN / 16, BATCH);      // 8 tiles x 64 batches = 512 waves
    att_score_kernel<<<grid, 32, 0, stream>>>(h, q, W_att, W_fc, s_ws, t_ws);
    att_softmax_pool_kernel<<<BATCH, PLEN, 0, stream>>>(s_ws, t_ws, b_fc, out);
}